// IndexSelecter_49503793054535
// MI455X (gfx1250) — compile-verified
//
#include <hip/hip_runtime.h>
#include <cstdint>

// Bicubic IndexSelecter for MI455X (gfx1250, wave32).
//   input (B=4, C=64, H=512, W=512) f32, index (B, K=65536, 2) f32 in (-0.99, 0.99)
//   output (B, K, C) f32
// Per-wave async Global->LDS gather pipeline (ASYNCcnt double buffer),
// branchless cubic weights, ds_load_b128 compute reads, NT output stores.

#define A_COEF (-0.75f)

constexpr int BB   = 4;
constexpr int CC   = 64;
constexpr int HH   = 512;
constexpr int WW   = 512;
constexpr int KK   = 65536;          // power of two: b = p >> 16
constexpr int NPTS = BB * KK;        // 262144 points
constexpr int HWs  = HH * WW;

typedef __attribute__((ext_vector_type(4))) float vfloat4;
typedef __attribute__((ext_vector_type(2))) float vfloat2;

// l1(s) for s in [0,1], l2(s) for s in [1,2] (reference's where(s<=1,l1,l2)).
// Both are exactly 0 at s==1, so per-tap static selection is exact.
__device__ __forceinline__ float cubic_l1(float s) {
  float s2 = s * s, s3 = s2 * s;
  return (A_COEF + 2.0f) * s3 - (A_COEF + 3.0f) * s2 + 1.0f;
}
__device__ __forceinline__ float cubic_l2(float s) {
  float s2 = s * s, s3 = s2 * s;
  return A_COEF * s3 - 5.0f * A_COEF * s2 + 8.0f * A_COEF * s - 4.0f * A_COEF;
}

// Branchless 4-tap weights for fractional offset t in [0,1):
// distances are 1+t, t, 1-t, 2-t -> l2, l1, l1, l2 statically.
__device__ __forceinline__ void cubic_weights(float t, float* w) {
  w[0] = cubic_l2(1.0f + t);
  w[1] = cubic_l1(t);
  w[2] = cubic_l1(1.0f - t);
  w[3] = cubic_l2(2.0f - t);
}

// Compute weights for point p and issue the 8 async b128 gathers for this
// lane's share of the 4x4xC patch into the LDS buffer at byte address ldsaddr.
// Lane j covers row l = j&3 of channels c = 8*i + (j>>2), i = 0..7.
__device__ __forceinline__ void point_setup(
    int p, const float* __restrict__ idx, const float* __restrict__ inp,
    int lane, uint32_t ldsaddr, float* wx, float* wy)
{
  const int b = p >> 16;                    // K == 65536
  const float gx = idx[2 * p + 0];
  const float gy = idx[2 * p + 1];
  const float x = (gx + 1.0f) * (0.5f * (float)(WW - 1));
  const float y = (gy + 1.0f) * (0.5f * (float)(HH - 1));
  const float fx = floorf(x), fy = floorf(y);
  const int ix0 = (int)fx - 1;
  const int iy0 = (int)fy - 1;
  cubic_weights(x - fx, wx);
  cubic_weights(y - fy, wy);
#pragma unroll
  for (int r = 0; r < 4; ++r) {
    // zeroing invalid taps' separable weights == reference's valid mask
    // (selects, not branches; never triggers for coords in (-0.99, 0.99))
    wx[r] = ((unsigned)(ix0 + r) < (unsigned)WW) ? wx[r] : 0.0f;
    wy[r] = ((unsigned)(iy0 + r) < (unsigned)HH) ? wy[r] : 0.0f;
  }
  const int bx  = min(max(ix0, 0), WW - 4);          // b128 4-tap row base
  const int l   = lane & 3;
  const int iyl = min(max(iy0 + l, 0), HH - 1);
  uint64_t addr = (uint64_t)(uintptr_t)inp
                + ((uint64_t)(b * CC + (lane >> 2)) * (uint64_t)HWs
                   + (uint64_t)(iyl * WW + bx)) * 4ull;
  uint32_t laddr = ldsaddr + (uint32_t)lane * 16u;   // element (c*4+l) = 32*i + lane

  // Drain our previous LDS reads before the async engine overwrites this buffer.
  asm volatile("s_wait_dscnt 0x0" ::: "memory");
#pragma unroll
  for (int i = 0; i < 8; ++i) {
    // per-lane 16B gather, lands in LDS; tracked by ASYNCcnt (in-order completion)
    asm volatile("global_load_async_to_lds_b128 %0, %1, off"
                 :: "v"(laddr), "v"(addr) : "memory");
    addr  += (uint64_t)(8 * HWs) * 4ull;   // +8 channels
    laddr += 512u;                          // 32 elements * 16B
  }
}

__global__ __launch_bounds__(256)
void bicubic_select_kernel(const float* __restrict__ inp,
                           const float* __restrict__ idx,
                           float* __restrict__ out,
                           int npoints, int nwaves)
{
  // 8 waves/block * 2 buffers * 256 float4 (4KB patch) = 64KB of 320KB WGP LDS
  __shared__ vfloat4 lds4[8 * 2 * 256];
  const int lane = (int)threadIdx.x & 31;
  const int wv   = (int)threadIdx.x >> 5;
  const int gw   = (int)blockIdx.x * 8 + wv;

  const unsigned bufbase = (unsigned)(wv * 2) * 256u;   // element index of buffer 0
  // low 32 bits of the flat shared address == LDS byte offset (ISA 10.2)
  const uint32_t lb0 = (uint32_t)(uintptr_t)&lds4[bufbase];
  const uint32_t lb1 = lb0 + 4096u;

  int p = gw;                      // wave-uniform point id
  if (p >= npoints) return;

  float wxc[4], wyc[4], wxn[4], wyn[4];
  point_setup(p, idx, inp, lane, lb0, wxc, wyc);
  unsigned d = 0;
  for (;;) {
    const int  pn = p + nwaves;
    const bool hn = pn < npoints;
    if (hn) {
      point_setup(pn, idx, inp, lane, d ? lb0 : lb1, wxn, wyn); // prefetch next
      asm volatile("s_wait_asynccnt 0x8" ::: "memory");  // current 8 done, next 8 in flight
    } else {
      asm volatile("s_wait_asynccnt 0x0" ::: "memory");
    }
    // Lane reduces channels 2*lane, 2*lane+1 straight from LDS (ds_load_b128:
    // direct integer indexing into lds4 keeps these in addrspace(3)).
    {
      const unsigned cb = bufbase + d * 256u;
      const int c0 = 2 * lane;
      float acc0 = 0.0f, acc1 = 0.0f;
#pragma unroll
      for (int l = 0; l < 4; ++l) {
        vfloat4 v0 = lds4[cb + (unsigned)(c0 * 4 + l)];
        vfloat4 v1 = lds4[cb + (unsigned)((c0 + 1) * 4 + l)];
        float d0 = v0.x * wxc[0] + v0.y * wxc[1] + v0.z * wxc[2] + v0.w * wxc[3];
        float d1 = v1.x * wxc[0] + v1.y * wxc[1] + v1.z * wxc[2] + v1.w * wxc[3];
        acc0 = fmaf(wyc[l], d0, acc0);
        acc1 = fmaf(wyc[l], d1, acc1);
      }
      vfloat2 o; o.x = acc0; o.y = acc1;
      // streaming output: keep L2 for the gathered input image
      __builtin_nontemporal_store(o, (vfloat2*)&out[(size_t)p * CC + c0]);
    }
    if (!hn) break;
    p = pn; d ^= 1u;
#pragma unroll
    for (int r = 0; r < 4; ++r) { wxc[r] = wxn[r]; wyc[r] = wyn[r]; }
  }
}

extern "C" void kernel_launch(void* const* d_in, const int* in_sizes, int n_in,
                              void* d_out, int out_size, void* d_ws, size_t ws_size,
                              hipStream_t stream) {
  (void)in_sizes; (void)n_in; (void)d_ws; (void)ws_size; (void)out_size;
  const float* inp = (const float*)d_in[0];   // (B, C, H, W) f32
  const float* idx = (const float*)d_in[1];   // (B, K, 2)  f32
  float* out = (float*)d_out;                 // (B, K, C)  f32

  const int blocks = 1024;                    // 8192 waves, 32 points each
  bicubic_select_kernel<<<blocks, 256, 0, stream>>>(inp, idx, out, NPTS, blocks * 8);
}